// BatchedFoveator_1185410974201
// MI455X (gfx1250) — compile-verified
//
#include <hip/hip_runtime.h>
#include <stdint.h>
#include <stddef.h>

// ---------------------------------------------------------------------------
// BatchedFoveator on MI455X (gfx1250).
// out[b,t,c,i,j] = (1/s^2) * sum_{dy<s,dx<s} img[b,c, y0+s*i+dy, x0+s*j+dx]
//
// Per (batch, token) workgroup (1 wave32):
//   - 3x per-channel TDM tensor_load_to_lds (async, TENSORcnt), staggered
//     s_wait_tensorcnt(2/1/0) so channel c computes while c+1, c+2 stream in.
//   - horizontal SxS pooling in LDS (per-lane vector adds)
//   - vertical pooling = P(16xR) * T(Rx16) via V_WMMA_F32_16X16X4_F32 with
//     0/1 pooling matrix P (exact in fp32), D scaled by 1/S^2, NT stores.
// ---------------------------------------------------------------------------

typedef __attribute__((ext_vector_type(2))) float         v2f;
typedef __attribute__((ext_vector_type(8))) float         v8f;
typedef __attribute__((ext_vector_type(4))) unsigned int  u32x4;
typedef __attribute__((ext_vector_type(4))) int           i32x4;
typedef __attribute__((ext_vector_type(8))) int           i32x8;

#define IMG        512
#define PLANE      (IMG * IMG)
#define NTOK_TOTAL 160

#if defined(__has_builtin)
#if __has_builtin(__builtin_amdgcn_tensor_load_to_lds)
#define HAVE_TDM 1
#endif
#endif

#ifdef HAVE_TDM
// D# per CDNA5 ISA: 2-D tile (x=tileW, y=tileH),
// element = 4 bytes, row stride = rowStride elements.
__device__ __forceinline__ void tdm_load_tile2d(uint32_t lds_off, uint64_t gaddr,
                                                int tileW, int tileH,
                                                uint32_t rowStride) {
  u32x4 g0;
  g0[0] = 1u;                                                  // count=1 (valid), user mode
  g0[1] = lds_off;                                             // lds_addr (bytes)
  g0[2] = (uint32_t)gaddr;                                     // global_addr[31:0]
  g0[3] = ((uint32_t)(gaddr >> 32) & 0x1FFFFFFu) | (2u << 30); // addr[56:32] | type=2

  i32x8 g1;
  g1[0] = 0x00020000;                                          // data_size=2 (4B)
  g1[1] = (int)((uint32_t)IMG << 16);                          // tensor_dim0[15:0]
  g1[2] = (int)((uint32_t)IMG << 16);                          // dim0 hi=0 | tensor_dim1[15:0]
  g1[3] = (int)((uint32_t)tileW << 16);                        // dim1 hi=0 | tile_dim0
  g1[4] = (int)(uint32_t)tileH;                                // tile_dim1 | tile_dim2=0
  g1[5] = (int)rowStride;                                      // tensor_dim0_stride[31:0]
  g1[6] = 0;                                                   // s0 hi | s1 lo = 0
  g1[7] = 0;                                                   // s1 hi = 0

  i32x4 g2 = {0, 0, 0, 0};
  i32x4 g3 = {0, 0, 0, 0};

#if __clang_major__ >= 23
  i32x8 g4 = {0, 0, 0, 0, 0, 0, 0, 0};
  __builtin_amdgcn_tensor_load_to_lds(g0, g1, g2, g3, g4, 0);
#else
  __builtin_amdgcn_tensor_load_to_lds(g0, g1, g2, g3, 0);
#endif
}
#endif

// Token-in-level -> (row i, col j) of the level's 8x8 grid, mirroring the
// reference enumeration (full grid for level 0; rt=2 ring for levels 1,2).
template <bool RING>
__device__ __forceinline__ void tok_ij(int tl, int& i, int& j) {
  if (!RING) { i = tl >> 3; j = tl & 7; return; }
  if (tl < 16)      { i = tl >> 3; j = tl & 7; }                       // rows 0..1
  else if (tl < 32) { int q = tl - 16; i = 2 + (q >> 2);
                      int jj = q & 3; j = (jj < 2) ? jj : jj + 4; }    // rows 2..5, cols {0,1,6,7}
  else              { int q = tl - 32; i = 6 + (q >> 3); j = q & 7; }  // rows 6..7
}

template <int S, int OFF, int TOKBASE, bool RING>
__global__ __launch_bounds__(32)
void fov_kernel(const float* __restrict__ img, float* __restrict__ out) {
  constexpr int R  = 16 * S;   // tile rows (y)
  constexpr int Cw = 16 * S;   // tile cols (x)

  extern __shared__ float smem[];
  float* raw  = smem;                 // [3][R][Cw] raw pixels
  float* Tbuf = smem + 3 * R * Cw;    // [R][16] horizontally pooled (S>1 only)

  const int lane = threadIdx.x;       // 0..31 (wave32)
  const int tl   = blockIdx.x;        // token within level
  const int b    = blockIdx.y;        // batch

  int ti, tj;
  tok_ij<RING>(tl, ti, tj);
  const int x0 = 16 * (OFF + S * tj);
  const int y0 = 16 * (OFF + S * ti);

  const float* src = img + (size_t)b * 3 * PLANE + (size_t)y0 * IMG + x0;

#ifdef HAVE_TDM
  // Issue one async TDM tile load per channel; process pipelined below.
  const uint32_t lds_base = (uint32_t)(uintptr_t)raw;
#pragma unroll
  for (int ch = 0; ch < 3; ++ch)
    tdm_load_tile2d(lds_base + (uint32_t)(ch * R * Cw * 4),
                    (uint64_t)(uintptr_t)(src + (size_t)ch * PLANE),
                    Cw, R, IMG);
#else
  for (int e = lane; e < 3 * R * Cw; e += 32) {
    int z = e / (R * Cw), rem = e % (R * Cw);
    int y = rem / Cw, x = rem % Cw;
    raw[e] = src[(size_t)z * PLANE + (size_t)y * IMG + x];
  }
  __syncthreads();
#endif

  const int   hi  = lane >> 4;      // half-wave select (K/M offset per ISA layout)
  const int   mn  = lane & 15;      // M row of A-fragment == N col of B/D-fragment
  const float inv_area = 1.0f / (float)(S * S);
  const int   t = TOKBASE + tl;
  float* obase = out + (((size_t)b * NTOK_TOTAL + t) * 3) * 256;

#pragma unroll
  for (int ch = 0; ch < 3; ++ch) {
#ifdef HAVE_TDM
    // Channel ch's tile is the (ch+1)-th oldest outstanding tensor op; tensor
    // ops of one wave complete in order, so wait TENSORcnt <= 2-ch.
    // (builtin requires a literal constant -> distinct call sites)
    if (ch == 0)      __builtin_amdgcn_s_wait_tensorcnt(2);
    else if (ch == 1) __builtin_amdgcn_s_wait_tensorcnt(1);
    else              __builtin_amdgcn_s_wait_tensorcnt(0);
    asm volatile("" ::: "memory");
#endif
    const float* plane = raw + ch * R * Cw;
    const float* Tp;

    if (S == 1) {
      Tp = plane;                   // no horizontal pooling needed
    } else {
      // Stage 1: T[r][j] = sum_{dx<S} X[r][S*j+dx]  (uniform trips, no divergence)
      for (int e = lane; e < R * 16; e += 32) {
        int r = e >> 4, j = e & 15;
        const float* p = plane + r * Cw + j * S;
        float ssum = 0.0f;
#pragma unroll
        for (int dx = 0; dx < S; ++dx) ssum += p[dx];
        Tbuf[r * 16 + j] = ssum;
      }
      __syncthreads();              // single wave: cheap; orders T for WMMA reads
      Tp = Tbuf;
    }

    // Stage 2: out = P (16 x R) * T (R x 16), P(i,c)=1 iff c/S==i, via f32 WMMA.
    // A 16x4 layout: lane m holds K = kb+2*hi (+1); B 4x16: lane n holds rows kb+2*hi (+1).
    v8f acc = {};
#pragma unroll
    for (int kb = 0; kb < R; kb += 4) {
      const int c0 = kb + 2 * hi;
      v2f a, bf;
      a.x = ((c0 / S) == mn) ? 1.0f : 0.0f;
      a.y = (((c0 + 1) / S) == mn) ? 1.0f : 0.0f;
      bf.x = Tp[c0 * 16 + mn];
      bf.y = Tp[(c0 + 1) * 16 + mn];
      acc = __builtin_amdgcn_wmma_f32_16x16x4_f32(
          /*neg_a=*/false, a, /*neg_b=*/false, bf,
          /*c_mod=*/(short)0, acc, /*reuse_a=*/false, /*reuse_b=*/false);
    }

    // D layout: VGPR r -> (M = r + 8*hi, N = mn). Streaming (non-temporal) stores.
    float* oc = obase + ch * 256;
#pragma unroll
    for (int r = 0; r < 8; ++r)
      __builtin_nontemporal_store(acc[r] * inv_area, oc + (r + 8 * hi) * 16 + mn);

    if (S > 1) __syncthreads();     // Tbuf reused next channel
  }
}

extern "C" void kernel_launch(void* const* d_in, const int* in_sizes, int n_in,
                              void* d_out, int out_size, void* d_ws, size_t ws_size,
                              hipStream_t stream) {
  const float* img = (const float*)d_in[0];
  float*       out = (float*)d_out;
  const int B = in_sizes[0] / (3 * PLANE);   // 32

  // Level 0: 64 tokens, S=1, block-offset 12
  {
    size_t lds = (size_t)3 * 16 * 16 * sizeof(float);
    fov_kernel<1, 12, 0, false><<<dim3(64, B), dim3(32), lds, stream>>>(img, out);
  }
  // Level 1: 48 tokens, S=2, block-offset 8
  {
    size_t lds = ((size_t)3 * 32 * 32 + 32 * 16) * sizeof(float);
    fov_kernel<2, 8, 64, true><<<dim3(48, B), dim3(32), lds, stream>>>(img, out);
  }
  // Level 2: 48 tokens, S=4, block-offset 0
  {
    size_t lds = ((size_t)3 * 64 * 64 + 64 * 16) * sizeof(float);
    fov_kernel<4, 0, 112, true><<<dim3(48, B), dim3(32), lds, stream>>>(img, out);
  }
}